// GSMonomialSSMLayer_62302795596494
// MI455X (gfx1250) — compile-verified
//
#include <hip/hip_runtime.h>
#include <hip/hip_bf16.h>
#include <math.h>

typedef float v2f __attribute__((ext_vector_type(2)));
typedef float v8f __attribute__((ext_vector_type(8)));
typedef unsigned u32x4 __attribute__((ext_vector_type(4)));
typedef int i32x4 __attribute__((ext_vector_type(4)));
typedef int i32x8 __attribute__((ext_vector_type(8)));

#define T_TOK   32768      // 8 * 4096
#define D_DIM   1024
#define SEQ     4096
#define NBATCH  8
#define NCHUNK  64         // chunks per sequence
#define CHUNKL  64         // steps per chunk  (64*64 = 4096)
#define TAU_INV 2.0f
#define EPS_RMS 1e-6f
#define XS_STRIDE 1028     // padded LDS row stride (floats): banks = 4*row+c, conflict-free

// ---------------------------------------------------------------------------
// fp32 WMMA: D(16x16) = A(16x4) * B(4x16) + C   -> v_wmma_f32_16x16x4_f32
// ---------------------------------------------------------------------------
__device__ __forceinline__ v8f wmma4(v2f a, v2f b, v8f c) {
  return __builtin_amdgcn_wmma_f32_16x16x4_f32(false, a, false, b, (short)0, c,
                                               false, false);
}

// ---------------------------------------------------------------------------
// TDM: DMA a 16-row x 1024-col f32 tile (row stride 1024) from global memory
// into LDS with padded row stride XS_STRIDE, via one iterating descriptor:
// 16 iterations x (1 row of 1024 f32), global +1024 elems, LDS +1028 elems.
// ---------------------------------------------------------------------------
__device__ __forceinline__ void tdm_load_x16(const float* gsrc,
                                             unsigned lds_off) {
  unsigned long long ga = (unsigned long long)(size_t)gsrc;
  u32x4 g0;
  g0[0] = 1u;                                   // count=1, user mode, no gather
  g0[1] = lds_off;                              // LDS byte address
  g0[2] = (unsigned)ga;                         // global_addr[31:0]
  g0[3] = (unsigned)(ga >> 32) | 0x80000000u;   // global_addr[56:32] | type=2

  i32x8 g1;
  g1[0] = (int)((2u << 16) | (1u << 19));       // data_size=4B, iterate_enable
  g1[1] = (int)(1024u << 16);                   // tensor_dim0[15:0]=1024
  g1[2] = (int)(16u << 16);                     // tensor_dim0 hi=0 | tensor_dim1[15:0]=16
  g1[3] = (int)(1024u << 16);                   // tensor_dim1 hi=0 | tile_dim0=1024
  g1[4] = 1;                                    // tile_dim1=1, tile_dim2=0
  g1[5] = 1024;                                 // tensor_dim0_stride lo
  g1[6] = (int)(1024u << 16);                   // stride hi=0 | tensor_dim1_stride lo
  g1[7] = 0;

  i32x4 g2;
  g2[0] = 1;                                    // tensor_dim2
  g2[1] = XS_STRIDE;                            // lds_addr_increment (elements)
  g2[2] = 1024;                                 // global_addr_increment (elements)
  g2[3] = (int)(15u << 16);                     // iterate_count = 15 -> 16 iters

  i32x4 g3 = {0, 0, 0, 0};

#if defined(__clang_major__) && (__clang_major__ >= 23)
  i32x8 z8 = {};
  __builtin_amdgcn_tensor_load_to_lds(g0, g1, g2, g3, z8, 0);
#else
  __builtin_amdgcn_tensor_load_to_lds(g0, g1, g2, g3, 0);
#endif
}

// ---------------------------------------------------------------------------
// K1: projections (WMMA GEMM, N=208) + sinkhorn/argmax/diag/b precompute
// ---------------------------------------------------------------------------
__device__ __forceinline__ const float* wrow_base(
    int nt, const float* WpR, const float* WdR, const float* WaR,
    const float* WpL, const float* WdL, const float* WaL, const float* WB) {
  if (nt < 4)  return WpR + (size_t)nt * 16 * D_DIM;
  if (nt == 4) return WdR;
  if (nt == 5) return WaR;
  if (nt < 10) return WpL + (size_t)(nt - 6) * 16 * D_DIM;
  if (nt == 10) return WdL;
  if (nt == 11) return WaL;
  return WB;
}

// sinkhorn(5 iters) + per-column first-argmax on one 4x4 block -> 8-bit code
__device__ __forceinline__ unsigned sinkhorn_block(const float* __restrict__ pl) {
  float la[4][4];
#pragma unroll
  for (int i = 0; i < 4; ++i)
#pragma unroll
    for (int j = 0; j < 4; ++j) la[i][j] = pl[i * 4 + j] * TAU_INV;
#pragma unroll
  for (int it = 0; it < 5; ++it) {
#pragma unroll
    for (int i = 0; i < 4; ++i) {  // row logsumexp (axis=-1)
      float m = fmaxf(fmaxf(la[i][0], la[i][1]), fmaxf(la[i][2], la[i][3]));
      float s = expf(la[i][0] - m) + expf(la[i][1] - m) + expf(la[i][2] - m) +
                expf(la[i][3] - m);
      float lse = m + logf(s);
#pragma unroll
      for (int j = 0; j < 4; ++j) la[i][j] -= lse;
    }
#pragma unroll
    for (int j = 0; j < 4; ++j) {  // col logsumexp (axis=-2)
      float m = fmaxf(fmaxf(la[0][j], la[1][j]), fmaxf(la[2][j], la[3][j]));
      float s = expf(la[0][j] - m) + expf(la[1][j] - m) + expf(la[2][j] - m) +
                expf(la[3][j] - m);
      float lse = m + logf(s);
#pragma unroll
      for (int i = 0; i < 4; ++i) la[i][j] -= lse;
    }
  }
  unsigned code = 0u;
#pragma unroll
  for (int j = 0; j < 4; ++j) {
    float best = la[0][j];
    int arg = 0;
#pragma unroll
    for (int i = 1; i < 4; ++i)
      if (la[i][j] > best) { best = la[i][j]; arg = i; }
    code |= (unsigned)arg << (2 * j);
  }
  return code;
}

__global__ void __launch_bounds__(256)
k_proj(const float* __restrict__ x, const float* __restrict__ mask,
       const float* __restrict__ WpR, const float* __restrict__ WdR,
       const float* __restrict__ WaR, const float* __restrict__ WpL,
       const float* __restrict__ WdL, const float* __restrict__ WaL,
       const float* __restrict__ WB, float* __restrict__ dR,
       float* __restrict__ dL, float* __restrict__ bt,
       unsigned* __restrict__ selR, unsigned* __restrict__ selL) {
  __shared__ __align__(16) float xs[16 * XS_STRIDE];
  __shared__ float proj_s[16][208];
  __shared__ unsigned char sel_s[16][2][4];
  const int lane = threadIdx.x & 31;
  const int wave = threadIdx.x >> 5;
  const size_t tok0 = (size_t)blockIdx.x * 16;

  if (threadIdx.x < 32) {  // wave 0 drives the tensor DMA, others park at barrier
    tdm_load_x16(x + tok0 * D_DIM, (unsigned)(size_t)(void*)xs);
    __builtin_amdgcn_s_wait_tensorcnt(0);
  }
  __syncthreads();

  // GEMM: 16 tokens x 208 outputs, K=1024, fp32 WMMA 16x16x4; A from LDS
  const int koff = (lane & 16) >> 3;
  const float* arow = &xs[(lane & 15) * XS_STRIDE + koff];
  for (int nt = wave; nt < 13; nt += 8) {
    const float* wb = wrow_base(nt, WpR, WdR, WaR, WpL, WdL, WaL, WB);
    const float* brow = wb + (size_t)(lane & 15) * D_DIM + koff;
    v8f acc = {};
#pragma unroll 4
    for (int k0 = 0; k0 < D_DIM; k0 += 4) {
      v2f a = *(const v2f*)(arow + k0);
      v2f b = *(const v2f*)(brow + k0);
      acc = wmma4(a, b, acc);
    }
    const int m0 = (lane & 16) >> 1;  // 0 or 8
#pragma unroll
    for (int v = 0; v < 8; ++v) proj_s[m0 + v][nt * 16 + (lane & 15)] = acc[v];
  }
  __syncthreads();

  // sinkhorn+argmax: 128 threads, one (token, factor, 4x4-block) each
  if (threadIdx.x < 128) {
    const int tokL = threadIdx.x >> 3;
    const int fac = (threadIdx.x >> 2) & 1;
    const int blk = threadIdx.x & 3;
    const float* pl = proj_s[tokL] + fac * 96 + blk * 16;
    sel_s[tokL][fac][blk] = (unsigned char)sinkhorn_block(pl);
  }
  // diag + b: 256 threads, one (token, state-index) each
  {
    const int tokL = threadIdx.x >> 4;
    const int i = threadIdx.x & 15;
    const size_t tok = tok0 + tokL;
    const float* row = proj_s[tokL];
    float aR = 1.0f / (1.0f + expf(-row[80 + i]));
    dR[tok * 16 + i] = aR * tanhf(row[64 + i]);
    float aL = 1.0f / (1.0f + expf(-row[176 + i]));
    dL[tok * 16 + i] = aL * tanhf(row[160 + i]);
    bt[tok * 16 + i] = row[192 + i] * mask[tok];
  }
  __syncthreads();
  if (threadIdx.x < 32) {  // assemble packed 2-bit selectors
    const int tokL = threadIdx.x >> 1;
    const int fac = threadIdx.x & 1;
    unsigned sel = (unsigned)sel_s[tokL][fac][0] |
                   ((unsigned)sel_s[tokL][fac][1] << 8) |
                   ((unsigned)sel_s[tokL][fac][2] << 16) |
                   ((unsigned)sel_s[tokL][fac][3] << 24);
    (fac ? selL : selR)[tok0 + tokL] = sel;
  }
}

// ---------------------------------------------------------------------------
// Scan machinery: state h (16 floats) held lane-wise in two mirrored 16-lane
// halves of a wave32. One step: h <- D_L P_L S D_R P_R h + b.
// ---------------------------------------------------------------------------
__device__ __forceinline__ float scan_step(float h, float dRv, float dLv,
                                           float btv, unsigned sRu,
                                           unsigned sLu, int lane) {
  const int li = lane & 15, half = lane & 16, il = li & 3, bb = li & 12;
  float acc = 0.0f;  // R-block apply: acc_i = sum_{j in blk, sel[j]==i} h[j]
#pragma unroll
  for (int jl = 0; jl < 4; ++jl) {
    int j = bb + jl;
    int s = (sRu >> (2 * j)) & 3;
    float hj = __shfl(h, half + j, 32);
    acc += (s == il) ? hj : 0.0f;
  }
  float h1 = dRv * acc;
  int pi = ((li & 3) << 2) | (li >> 2);  // stride-perm gather (involution)
  float h2 = __shfl(h1, half + pi, 32);
  acc = 0.0f;
#pragma unroll
  for (int jl = 0; jl < 4; ++jl) {
    int j = bb + jl;
    int s = (sLu >> (2 * j)) & 3;
    float hj = __shfl(h2, half + j, 32);
    acc += (s == il) ? hj : 0.0f;
  }
  return dLv * acc + btv;
}

// step transform as column-monomial map: column li -> (row rt, value vt)
__device__ __forceinline__ void step_rv(float dRv, float dLv, unsigned sRu,
                                        unsigned sLu, int lane, int& rt,
                                        float& vt) {
  const int li = lane & 15, half = lane & 16;
  int row1 = (li & 12) | ((sRu >> (2 * li)) & 3);
  float dr = __shfl(dRv, half + row1, 32);
  int row2 = ((row1 & 3) << 2) | (row1 >> 2);
  int row3 = (row2 & 12) | ((sLu >> (2 * row2)) & 3);
  float dl = __shfl(dLv, half + row3, 32);
  rt = row3;
  vt = dr * dl;
}

// K2a: per-chunk sweep -> column-monomial chunk summary (R,V,c)
__global__ void __launch_bounds__(32)
k_chunk(const float* __restrict__ dR, const float* __restrict__ dL,
        const float* __restrict__ bt, const unsigned* __restrict__ selR,
        const unsigned* __restrict__ selL, unsigned* __restrict__ chR,
        float* __restrict__ chV, float* __restrict__ chC) {
  const int lane = threadIdx.x, li = lane & 15, half = lane & 16;
  const int b = blockIdx.x / NCHUNK, ch = blockIdx.x % NCHUNK;
  int Rv = li;
  float Vv = 1.0f, cv = 0.0f;
  size_t tok = (size_t)b * SEQ + (size_t)ch * CHUNKL;
  for (int s = 0; s < CHUNKL; ++s, ++tok) {
    float dRv = dR[tok * 16 + li], dLv = dL[tok * 16 + li],
          btv = bt[tok * 16 + li];
    unsigned sRu = selR[tok], sLu = selL[tok];
    int rt; float vt;
    step_rv(dRv, dLv, sRu, sLu, lane, rt, vt);
    int newR = __shfl(rt, half + Rv, 32);
    float newV = __shfl(vt, half + Rv, 32) * Vv;
    Rv = newR; Vv = newV;
    cv = scan_step(cv, dRv, dLv, btv, sRu, sLu, lane);
  }
  if (lane < 16) {
    size_t o = (size_t)blockIdx.x * 16 + li;
    chR[o] = (unsigned)Rv; chV[o] = Vv; chC[o] = cv;
  }
}

// K2b: sequential prefix over chunk summaries -> h_start per chunk
__global__ void __launch_bounds__(32)
k_prefix(const unsigned* __restrict__ chR, const float* __restrict__ chV,
         const float* __restrict__ chC, float* __restrict__ hst) {
  const int lane = threadIdx.x, li = lane & 15, half = lane & 16;
  const int b = blockIdx.x;
  float h = 0.0f;
  for (int k = 0; k < NCHUNK; ++k) {
    size_t o = ((size_t)b * NCHUNK + k) * 16;
    if (lane < 16) hst[o + li] = h;
    int Rk = (int)chR[o + li];
    float t = chV[o + li] * h;
    float nh = chC[o + li];
#pragma unroll
    for (int j = 0; j < 16; ++j) {  // (A h)[i] = sum_{j: R[j]==i} V[j] h[j]
      int rj = __shfl(Rk, half + j, 32);
      float tj = __shfl(t, half + j, 32);
      nh += (rj == li) ? tj : 0.0f;
    }
    h = nh;
  }
}

// K2c: replay each chunk from its h_start, emit hs
__global__ void __launch_bounds__(32)
k_replay(const float* __restrict__ dR, const float* __restrict__ dL,
         const float* __restrict__ bt, const unsigned* __restrict__ selR,
         const unsigned* __restrict__ selL, const float* __restrict__ hst,
         float* __restrict__ hs) {
  const int lane = threadIdx.x, li = lane & 15;
  const int b = blockIdx.x / NCHUNK, ch = blockIdx.x % NCHUNK;
  float h = hst[(size_t)blockIdx.x * 16 + li];
  size_t tok = (size_t)b * SEQ + (size_t)ch * CHUNKL;
  for (int s = 0; s < CHUNKL; ++s, ++tok) {
    h = scan_step(h, dR[tok * 16 + li], dL[tok * 16 + li], bt[tok * 16 + li],
                  selR[tok], selL[tok], lane);
    if (lane < 16) hs[tok * 16 + li] = h;
  }
}

// ---------------------------------------------------------------------------
// K3: y = sigmoid(x @ Wg^T) * (hs @ Wc^T), RMS-norm, *norm_w   (fp32 WMMA)
// ---------------------------------------------------------------------------
__global__ void __launch_bounds__(256)
k_out(const float* __restrict__ x, const float* __restrict__ hs,
      const float* __restrict__ Wg, const float* __restrict__ WC,
      const float* __restrict__ nw, float* __restrict__ out) {
  __shared__ __align__(16) float xs[16 * XS_STRIDE];
  __shared__ float red[8][16];
  __shared__ float rinv_s[16];
  const int lane = threadIdx.x & 31;
  const int w = threadIdx.x >> 5;
  const size_t tok0 = (size_t)blockIdx.x * 16;
  const int koff = (lane & 16) >> 3;

  if (threadIdx.x < 32) {  // wave 0 drives the tensor DMA for the x tile
    tdm_load_x16(x + tok0 * D_DIM, (unsigned)(size_t)(void*)xs);
    __builtin_amdgcn_s_wait_tensorcnt(0);
  }
  __syncthreads();

  v8f acc[8];
#pragma unroll
  for (int q = 0; q < 8; ++q) acc[q] = (v8f){};

  const float* arow = &xs[(lane & 15) * XS_STRIDE + koff];
#pragma unroll 2
  for (int k0 = 0; k0 < D_DIM; k0 += 4) {
    v2f a = *(const v2f*)(arow + k0);
#pragma unroll
    for (int q = 0; q < 8; ++q) {
      int e = (w * 8 + q) * 16 + (lane & 15);
      v2f bf = *(const v2f*)(Wg + (size_t)e * D_DIM + k0 + koff);
      acc[q] = wmma4(a, bf, acc[q]);
    }
  }

  // fuse: gate = sigmoid(acc); y = gate * (hs @ Wc^T); accumulate y^2
  float p[8];
#pragma unroll
  for (int v = 0; v < 8; ++v) p[v] = 0.0f;
  const float* hrow = hs + (tok0 + (lane & 15)) * 16 + koff;
#pragma unroll
  for (int q = 0; q < 8; ++q) {
    int e = (w * 8 + q) * 16 + (lane & 15);
    v8f h2 = {};
    const float* crow = WC + (size_t)e * 16 + koff;
#pragma unroll
    for (int kk = 0; kk < 16; kk += 4) {
      v2f a2 = *(const v2f*)(hrow + kk);
      v2f b2 = *(const v2f*)(crow + kk);
      h2 = wmma4(a2, b2, h2);
    }
#pragma unroll
    for (int v = 0; v < 8; ++v) {
      float g = 1.0f / (1.0f + expf(-acc[q][v]));
      float val = g * h2[v];
      acc[q][v] = val;
      p[v] += val * val;
    }
  }

  // reduce y^2 across the 16 lanes of each half (the e-dimension of a tile)
#pragma unroll
  for (int v = 0; v < 8; ++v) {
    p[v] += __shfl_xor(p[v], 1, 32);
    p[v] += __shfl_xor(p[v], 2, 32);
    p[v] += __shfl_xor(p[v], 4, 32);
    p[v] += __shfl_xor(p[v], 8, 32);
  }
  if ((lane & 15) == 0) {
    const int m0 = (lane & 16) >> 1;
#pragma unroll
    for (int v = 0; v < 8; ++v) red[w][m0 + v] = p[v];
  }
  __syncthreads();
  if (threadIdx.x < 16) {
    float ss = 0.0f;
#pragma unroll
    for (int wv = 0; wv < 8; ++wv) ss += red[wv][threadIdx.x];
    rinv_s[threadIdx.x] = 1.0f / sqrtf(ss * (1.0f / (float)D_DIM) + EPS_RMS);
  }
  __syncthreads();

  const int m0 = (lane & 16) >> 1;
#pragma unroll
  for (int q = 0; q < 8; ++q) {
    int e = (w * 8 + q) * 16 + (lane & 15);
    float nwe = nw[e];
#pragma unroll
    for (int v = 0; v < 8; ++v) {
      int m = m0 + v;
      out[(tok0 + m) * D_DIM + e] = acc[q][v] * rinv_s[m] * nwe;
    }
  }
}

// ---------------------------------------------------------------------------
extern "C" void kernel_launch(void* const* d_in, const int* in_sizes, int n_in,
                              void* d_out, int out_size, void* d_ws,
                              size_t ws_size, hipStream_t stream) {
  const float* x    = (const float*)d_in[0];
  const float* mask = (const float*)d_in[1];
  const float* WpR  = (const float*)d_in[2];
  const float* WdR  = (const float*)d_in[3];
  const float* WaR  = (const float*)d_in[4];
  const float* WpL  = (const float*)d_in[5];
  const float* WdL  = (const float*)d_in[6];
  const float* WaL  = (const float*)d_in[7];
  const float* WB   = (const float*)d_in[8];
  const float* WC   = (const float*)d_in[9];
  const float* Wg   = (const float*)d_in[10];
  const float* nw   = (const float*)d_in[11];
  // d_in[12] shuffle_perm is the fixed stride permutation (involution), hardcoded.
  float* outp = (float*)d_out;

  // workspace layout (~8.8 MB)
  float* wsf = (float*)d_ws;
  float* dR  = wsf;                        // T*16
  float* dL  = dR + (size_t)T_TOK * 16;
  float* bt  = dL + (size_t)T_TOK * 16;
  float* hsb = bt + (size_t)T_TOK * 16;    // hs: T*16
  float* chV = hsb + (size_t)T_TOK * 16;   // 8*64*16
  float* chC = chV + NBATCH * NCHUNK * 16;
  float* hst = chC + NBATCH * NCHUNK * 16;
  unsigned* selR = (unsigned*)(hst + NBATCH * NCHUNK * 16);  // T
  unsigned* selL = selR + T_TOK;
  unsigned* chR  = selL + T_TOK;           // 8*64*16

  k_proj<<<T_TOK / 16, 256, 0, stream>>>(x, mask, WpR, WdR, WaR, WpL, WdL, WaL,
                                         WB, dR, dL, bt, selR, selL);
  k_chunk<<<NBATCH * NCHUNK, 32, 0, stream>>>(dR, dL, bt, selR, selL, chR, chV,
                                              chC);
  k_prefix<<<NBATCH, 32, 0, stream>>>(chR, chV, chC, hst);
  k_replay<<<NBATCH * NCHUNK, 32, 0, stream>>>(dR, dL, bt, selR, selL, hst,
                                               hsb);
  k_out<<<T_TOK / 16, 256, 0, stream>>>(x, hsb, Wg, WC, nw, outp);
}